// SelfAttention_74620761800711
// MI455X (gfx1250) — compile-verified
//
#include <hip/hip_runtime.h>

#define DIM 2048
#define BATCH 64

typedef float v2f __attribute__((ext_vector_type(2)));
typedef float v8f __attribute__((ext_vector_type(8)));

#if __has_builtin(__builtin_amdgcn_exp2f)
#define FAST_EXP2(x) __builtin_amdgcn_exp2f(x)   // v_exp_f32
#else
#define FAST_EXP2(x) exp2f(x)
#endif

// ---------------------------------------------------------------------------
// Kernel 1: fused QKV GEMM  (y = x @ W.T), f32 WMMA 16x16x4.
// grid.x = 3 * 128 (matrix id * n-tiles), block = 256 threads (8 waves).
// Each wave owns a K-slice of 256 and accumulates the whole 64x16 tile
// (4 M-tiles). Cross-wave reduce via LDS slabs in a fixed order
// (deterministic; no fp atomics).
//
// f32 WMMA lane layout (ISA 7.12.2):
//   A 16x4:  lane L<16 -> M=L,   VGPR{0,1} = K{0,1};  lane L>=16 -> M=L-16, K{2,3}
//   B 4x16:  lane L<16 -> N=L,   VGPR{0,1} = K{0,1};  lane L>=16 -> N=L-16, K{2,3}
//   C 16x16: VGPR r, lane L -> M = r + 8*(L/16), N = L%16
// => both A and B are contiguous float2 loads per lane.
// ---------------------------------------------------------------------------
__global__ __launch_bounds__(256) void qkv_gemm_kernel(
    const float* __restrict__ x,
    const float* __restrict__ Wq,
    const float* __restrict__ Wk,
    const float* __restrict__ Wv,
    float* __restrict__ qkv /* [3][BATCH][DIM] */)
{
    const int nt  = blockIdx.x & 127;   // n-tile (16 output columns)
    const int mat = blockIdx.x >> 7;    // 0=q, 1=k, 2=v
    const float* __restrict__ W = (mat == 0) ? Wq : (mat == 1) ? Wk : Wv;
    float* __restrict__ out = qkv + mat * (BATCH * DIM);

    const int tid   = threadIdx.x;
    const int wave  = tid >> 5;
    const int lane  = tid & 31;
    const int n0    = nt * 16;
    const int mrow  = lane & 15;            // M (for A) / N (for B) within tile
    const int khalf = (lane >> 4) * 2;      // lane-local K sub-offset

    v8f acc0 = 0.0f, acc1 = 0.0f, acc2 = 0.0f, acc3 = 0.0f;

    const int kbeg = wave * (DIM / 8);      // 256-wide K slice per wave
    const int kend = kbeg + (DIM / 8);
    const float* __restrict__ brow = W + (size_t)(n0 + mrow) * DIM;
    const float* __restrict__ a0   = x + (size_t)(0 * 16 + mrow) * DIM;
    const float* __restrict__ a1   = x + (size_t)(1 * 16 + mrow) * DIM;
    const float* __restrict__ a2   = x + (size_t)(2 * 16 + mrow) * DIM;
    const float* __restrict__ a3   = x + (size_t)(3 * 16 + mrow) * DIM;

    for (int k = kbeg; k < kend; k += 4) {
        const int ko = k + khalf;
        v2f b   = *(const v2f*)(brow + ko);
        v2f va0 = *(const v2f*)(a0 + ko);
        v2f va1 = *(const v2f*)(a1 + ko);
        v2f va2 = *(const v2f*)(a2 + ko);
        v2f va3 = *(const v2f*)(a3 + ko);
        acc0 = __builtin_amdgcn_wmma_f32_16x16x4_f32(false, va0, false, b, (short)0, acc0, false, false);
        acc1 = __builtin_amdgcn_wmma_f32_16x16x4_f32(false, va1, false, b, (short)0, acc1, false, false);
        acc2 = __builtin_amdgcn_wmma_f32_16x16x4_f32(false, va2, false, b, (short)0, acc2, false, false);
        acc3 = __builtin_amdgcn_wmma_f32_16x16x4_f32(false, va3, false, b, (short)0, acc3, false, false);
    }

    // Deterministic cross-wave reduction: each wave dumps its 64x16 partial
    // tile to its own LDS slab, then each thread sums the 8 slabs in order.
    __shared__ float red[8][1024];
#pragma unroll
    for (int r = 0; r < 8; ++r) {
        red[wave][(0 * 8 + r) * 32 + lane] = acc0[r];
        red[wave][(1 * 8 + r) * 32 + lane] = acc1[r];
        red[wave][(2 * 8 + r) * 32 + lane] = acc2[r];
        red[wave][(3 * 8 + r) * 32 + lane] = acc3[r];
    }
    __syncthreads();

    for (int c = tid; c < 1024; c += 256) {
        float s = 0.0f;
#pragma unroll
        for (int w = 0; w < 8; ++w) s += red[w][c];
        const int p  = c >> 5;          // (m, r) pair
        const int ln = c & 31;          // original lane
        const int m  = p >> 3;
        const int r  = p & 7;
        const int row = m * 16 + r + 8 * (ln >> 4);   // batch index 0..63
        const int col = n0 + (ln & 15);
        out[(size_t)row * DIM + col] = s;
    }
}

// ---------------------------------------------------------------------------
// Kernel 2: rank-1 softmax attention.
// logits[b,i,j] = q[b,i]*k[b,j]  =>  row max = max(q_i*kmax_b, q_i*kmin_b).
// grid = (DIM/256, BATCH), block = 256. k[b,:], v[b,:] staged in LDS
// (all lanes read the same j -> LDS broadcast, conflict-free).
// ---------------------------------------------------------------------------
__global__ __launch_bounds__(256) void attn_kernel(
    const float* __restrict__ qkv, float* __restrict__ out)
{
    const float* __restrict__ q = qkv;
    const float* __restrict__ k = qkv + (size_t)BATCH * DIM;
    const float* __restrict__ v = qkv + (size_t)2 * BATCH * DIM;

    const int b   = blockIdx.y;
    const int tid = threadIdx.x;

    __shared__ float ks[DIM];
    __shared__ float vs[DIM];
    __shared__ float wmax[8];
    __shared__ float wmin[8];

    for (int j = tid; j < DIM; j += 256) {
        ks[j] = k[(size_t)b * DIM + j];
        vs[j] = v[(size_t)b * DIM + j];
    }
    __syncthreads();

    // per-batch kmax / kmin (wave32 shuffle tree + tiny LDS combine)
    float mx = -__builtin_inff();
    float mn =  __builtin_inff();
    for (int j = tid; j < DIM; j += 256) {
        mx = fmaxf(mx, ks[j]);
        mn = fminf(mn, ks[j]);
    }
#pragma unroll
    for (int o = 16; o >= 1; o >>= 1) {
        mx = fmaxf(mx, __shfl_xor(mx, o, 32));
        mn = fminf(mn, __shfl_xor(mn, o, 32));
    }
    if ((tid & 31) == 0) { wmax[tid >> 5] = mx; wmin[tid >> 5] = mn; }
    __syncthreads();
    mx = wmax[0]; mn = wmin[0];
#pragma unroll
    for (int w = 1; w < 8; ++w) {
        mx = fmaxf(mx, wmax[w]);
        mn = fminf(mn, wmin[w]);
    }

    const int   i   = blockIdx.x * 256 + tid;
    const float s   = q[(size_t)b * DIM + i];
    const float L2E = 1.44269504088896340736f;
    const float sc  = s * L2E;
    const float mb  = fmaxf(s * mx, s * mn) * L2E;   // row max in log2 domain

    float num = 0.0f, den = 0.0f;
#pragma unroll 4
    for (int j = 0; j < DIM; ++j) {
        const float e = FAST_EXP2(fmaf(sc, ks[j], -mb));  // v_exp_f32
        den += e;
        num  = fmaf(e, vs[j], num);
    }
    out[(size_t)b * DIM + i] = num / den;
}

// ---------------------------------------------------------------------------
extern "C" void kernel_launch(void* const* d_in, const int* in_sizes, int n_in,
                              void* d_out, int out_size, void* d_ws, size_t ws_size,
                              hipStream_t stream) {
    const float* x  = (const float*)d_in[0];
    const float* Wq = (const float*)d_in[1];
    const float* Wk = (const float*)d_in[2];
    const float* Wv = (const float*)d_in[3];
    float* qkv = (float*)d_ws;          // [3][BATCH][DIM] f32 = 1.5 MiB
    float* out = (float*)d_out;         // [BATCH][DIM] f32

    qkv_gemm_kernel<<<dim3(3 * 128), dim3(256), 0, stream>>>(x, Wq, Wk, Wv, qkv);
    attn_kernel<<<dim3(DIM / 256, BATCH), dim3(256), 0, stream>>>(qkv, out);
}